// Mamba_24378234372522
// MI455X (gfx1250) — compile-verified
//
#include <hip/hip_runtime.h>
#include <hip/hip_bf16.h>
#include <math.h>

#define D_MODEL 1024
#define VOCAB   32000
#define D_STATE 16
#define D_INNER 2048
#define DT_RANK 64
#define D_CONV  4
#define NLAYER  2
#define BB      2
#define LL      1024
#define BL      (BB * LL)   // 2048 rows

typedef float v2f __attribute__((ext_vector_type(2)));
typedef float v8f __attribute__((ext_vector_type(8)));

__device__ __forceinline__ float siluf(float x)     { return x / (1.0f + __expf(-x)); }
__device__ __forceinline__ float softplusf(float x) { return (x > 20.0f) ? x : log1pf(__expf(x)); }

__device__ __forceinline__ v8f wmma_f32(v2f a, v2f b, v8f c) {
    return __builtin_amdgcn_wmma_f32_16x16x4_f32(false, a, false, b, (short)0, c, false, false);
}

// ---------------------------------------------------------------------------
// C[M,N] = A[M,K] (row-major, lda) * W[N,K]^T  (W row-major NxK, contiguous)
// EPI: 0 = none, 1 = softplus, 2 = += Cin (residual, leading dim ldc)
// MT:  16-row tiles per wave (wave covers MT*16 rows)
// NT:  16-col tiles per wave (waves own only FULL tiles)
// Per K-step of 4: MT A-frags x NT B-frags -> MT*NT WMMAs from MT+NT loads,
// next step's fragments prefetched (software pipeline).
// Requirements: M % (16*MT) == 0, K % 4 == 0, grid sized so each wave's tiles
// are fully inside N (or the whole wave exits).
// ---------------------------------------------------------------------------
template <int EPI, int MT, int NT>
__global__ __launch_bounds__(256)
void gemm_wmma_f32(const float* __restrict__ A, const float* __restrict__ W,
                   const float* __restrict__ Cin, float* __restrict__ C,
                   int M, int N, int K, int lda, int ldc) {
    const int lane  = threadIdx.x & 31;
    const int wave  = threadIdx.x >> 5;
    const int h     = lane >> 4;    // half-wave: K pair (2h, 2h+1)
    const int lid   = lane & 15;    // row (A) / col (B,C,D) within 16-tile
    const int m0    = blockIdx.y * (16 * MT);
    const int nbase = (blockIdx.x * 8 + wave) * (NT * 16);
    if (nbase >= N) return;         // wave-uniform exit; EXEC stays all-ones

    v8f acc[MT][NT];
#pragma unroll
    for (int i = 0; i < MT; ++i)
#pragma unroll
        for (int t = 0; t < NT; ++t) {
            v8f z = {0.f, 0.f, 0.f, 0.f, 0.f, 0.f, 0.f, 0.f};
            acc[i][t] = z;
        }

    const float* aP[MT];
#pragma unroll
    for (int i = 0; i < MT; ++i)
        aP[i] = A + (size_t)(m0 + 16 * i + lid) * lda + 2 * h;
    const float* wp = W + (size_t)(nbase + lid) * K + 2 * h;

    // prologue: fragments for k0 = 0
    v2f a[MT], b[NT];
#pragma unroll
    for (int i = 0; i < MT; ++i) a[i] = *(const v2f*)(aP[i]);
#pragma unroll
    for (int t = 0; t < NT; ++t) b[t] = *(const v2f*)(wp + (size_t)(16 * t) * K);

    int k0 = 0;
    for (; k0 + 4 < K; k0 += 4) {
        // prefetch next K-step fragments while the matrix pipe is busy
        v2f na[MT], nb[NT];
#pragma unroll
        for (int i = 0; i < MT; ++i) na[i] = *(const v2f*)(aP[i] + k0 + 4);
#pragma unroll
        for (int t = 0; t < NT; ++t) nb[t] = *(const v2f*)(wp + (size_t)(16 * t) * K + k0 + 4);

#pragma unroll
        for (int t = 0; t < NT; ++t)
#pragma unroll
            for (int i = 0; i < MT; ++i)
                acc[i][t] = wmma_f32(a[i], b[t], acc[i][t]);

#pragma unroll
        for (int i = 0; i < MT; ++i) a[i] = na[i];
#pragma unroll
        for (int t = 0; t < NT; ++t) b[t] = nb[t];
    }
    // drain: last K-step
#pragma unroll
    for (int t = 0; t < NT; ++t)
#pragma unroll
        for (int i = 0; i < MT; ++i)
            acc[i][t] = wmma_f32(a[i], b[t], acc[i][t]);

    // C/D layout: VGPR r, half h, lane lid -> row (r + 8h), col lid
#pragma unroll
    for (int t = 0; t < NT; ++t) {
        const int n = nbase + 16 * t + lid;
#pragma unroll
        for (int i = 0; i < MT; ++i) {
#pragma unroll
            for (int r = 0; r < 8; ++r) {
                const int m = m0 + 16 * i + r + 8 * h;
                float v = acc[i][t][r];
                if (EPI == 1) v = softplusf(v);
                if (EPI == 2) v += Cin[(size_t)m * ldc + n];
                C[(size_t)m * ldc + n] = v;
            }
        }
    }
}

// ---------------------------------------------------------------------------
// x[row,:] = emb[ids[row],:]
// ---------------------------------------------------------------------------
__global__ void embed_kernel(const int* __restrict__ ids,
                             const float* __restrict__ emb,
                             float* __restrict__ x) {
    const int row = blockIdx.x;
    const int id  = ids[row];
    const float4* src = (const float4*)(emb + (size_t)id * D_MODEL);
    float4*       dst = (float4*)(x + (size_t)row * D_MODEL);
    for (int i = threadIdx.x; i < D_MODEL / 4; i += blockDim.x) dst[i] = src[i];
}

// ---------------------------------------------------------------------------
// RMSNorm over D_MODEL, one block (256 thr) per row
// ---------------------------------------------------------------------------
__global__ __launch_bounds__(256)
void rmsnorm_kernel(const float* __restrict__ x, const float* __restrict__ w,
                    float* __restrict__ o) {
    __shared__ float red[8];
    const int row = blockIdx.x;
    const float* xr = x + (size_t)row * D_MODEL;
    float s = 0.0f;
    for (int i = threadIdx.x; i < D_MODEL; i += 256) { float v = xr[i]; s += v * v; }
    for (int off = 16; off > 0; off >>= 1) s += __shfl_down(s, off, 32);
    if ((threadIdx.x & 31) == 0) red[threadIdx.x >> 5] = s;
    __syncthreads();
    if (threadIdx.x == 0) {
        float t = 0.0f;
        for (int i = 0; i < 8; ++i) t += red[i];
        red[0] = rsqrtf(t / (float)D_MODEL + 1e-5f);
    }
    __syncthreads();
    const float rs = red[0];
    for (int i = threadIdx.x; i < D_MODEL; i += 256)
        o[(size_t)row * D_MODEL + i] = xr[i] * rs * w[i];
}

// ---------------------------------------------------------------------------
// Causal depthwise conv (width 4) over L, + bias, then SiLU. One thread/elem.
// ---------------------------------------------------------------------------
__global__ __launch_bounds__(256)
void conv_silu_kernel(const float* __restrict__ xi, const float* __restrict__ cw,
                      const float* __restrict__ cb, float* __restrict__ u) {
    const int idx = blockIdx.x * 256 + threadIdx.x;
    if (idx >= BL * D_INNER) return;
    const int d  = idx & (D_INNER - 1);
    const int bl = idx >> 11;           // row index in [0, BL)
    const int l  = bl & (LL - 1);
    float acc = cb[d];
#pragma unroll
    for (int j = 0; j < D_CONV; ++j) {
        const int ls = l - (D_CONV - 1) + j;
        if (ls >= 0)
            acc += cw[d * D_CONV + j] * xi[(size_t)(bl - (D_CONV - 1) + j) * D_INNER + d];
    }
    u[idx] = siluf(acc);
}

// ---------------------------------------------------------------------------
// Selective scan. Thread = one (b,d) channel; 16-state in registers.
// Per step, B_t/C_t (16+16 floats, shared by all 2048 channels) staged in LDS.
// Writes yact = (scan_y + u*Dv) * silu(res).
// ---------------------------------------------------------------------------
__global__ __launch_bounds__(256)
void scan_kernel(const float* __restrict__ u, const float* __restrict__ delta,
                 const float* __restrict__ xdbl, const float* __restrict__ A_log,
                 const float* __restrict__ Dp, const float* __restrict__ res,
                 float* __restrict__ yact) {
    __shared__ float sB[D_STATE];
    __shared__ float sC[D_STATE];
    const int b = blockIdx.y;
    const int d = blockIdx.x * 256 + threadIdx.x;

    float Arow[D_STATE];
#pragma unroll
    for (int n = 0; n < D_STATE; ++n) Arow[n] = -__expf(A_log[(size_t)d * D_STATE + n]);
    const float Dv = Dp[d];

    float st[D_STATE];
#pragma unroll
    for (int n = 0; n < D_STATE; ++n) st[n] = 0.0f;

    for (int l = 0; l < LL; ++l) {
        const size_t row = (size_t)(b * LL + l);
        if (threadIdx.x < 32) {
            const int n = threadIdx.x & 15;
            const float v = xdbl[row * 96 + DT_RANK + (threadIdx.x < 16 ? 0 : D_STATE) + n];
            if (threadIdx.x < 16) sB[n] = v; else sC[n] = v;
        }
        __syncthreads();
        const float dt = delta[row * D_INNER + d];
        const float ut = u[row * D_INNER + d];
        const float du = dt * ut;
        float y = 0.0f;
#pragma unroll
        for (int n = 0; n < D_STATE; ++n) {
            st[n] = __expf(dt * Arow[n]) * st[n] + du * sB[n];
            y += st[n] * sC[n];
        }
        const float r = res[row * D_INNER + d];
        yact[row * D_INNER + d] = (y + ut * Dv) * siluf(r);
        __syncthreads();
    }
}

// ---------------------------------------------------------------------------
extern "C" void kernel_launch(void* const* d_in, const int* in_sizes, int n_in,
                              void* d_out, int out_size, void* d_ws, size_t ws_size,
                              hipStream_t stream) {
    const int*   ids      = (const int*)  d_in[0];
    const float* emb      = (const float*)d_in[1];
    const float* norm_f_w = (const float*)d_in[2];
    const float* W_in_l   = (const float*)d_in[3];
    const float* W_in_r   = (const float*)d_in[4];
    const float* conv_w   = (const float*)d_in[5];
    const float* conv_b   = (const float*)d_in[6];
    const float* W_x      = (const float*)d_in[7];
    const float* W_dt     = (const float*)d_in[8];
    const float* A_log    = (const float*)d_in[9];
    const float* Dp       = (const float*)d_in[10];
    const float* W_out    = (const float*)d_in[11];
    const float* norm_w   = (const float*)d_in[12];
    float* out = (float*)d_out;

    // workspace layout (floats)
    float* ws    = (float*)d_ws;
    float* x     = ws;                                   // BL * D_MODEL
    float* xn    = x     + (size_t)BL * D_MODEL;         // BL * D_MODEL
    float* res   = xn    + (size_t)BL * D_MODEL;         // BL * D_INNER
    float* xi    = res   + (size_t)BL * D_INNER;         // BL * D_INNER
    float* u     = xi    + (size_t)BL * D_INNER;         // BL * D_INNER
    float* xdbl  = u     + (size_t)BL * D_INNER;         // BL * 96
    float* delta = xdbl  + (size_t)BL * 96;              // BL * D_INNER
    float* yact  = delta + (size_t)BL * D_INNER;         // BL * D_INNER

    const dim3 blk(256);
    // MT=4, NT=4: wave = 64x64 tile, block = 64 rows x 512 cols
    auto grd44 = [](int M, int N) { return dim3((unsigned)((N + 511) / 512), (unsigned)(M / 64)); };
    // MT=4, NT=2: wave = 64x32 tile, block = 64 rows x 256 cols (N=96 case)
    auto grd42 = [](int M, int N) { return dim3((unsigned)((N + 255) / 256), (unsigned)(M / 64)); };

    embed_kernel<<<BL, 256, 0, stream>>>(ids, emb, x);

    for (int i = 0; i < NLAYER; ++i) {
        const float* Wl  = W_in_l + (size_t)i * D_INNER * D_MODEL;
        const float* Wr  = W_in_r + (size_t)i * D_INNER * D_MODEL;
        const float* cw  = conv_w + (size_t)i * D_INNER * D_CONV;
        const float* cb  = conv_b + (size_t)i * D_INNER;
        const float* Wx  = W_x    + (size_t)i * 96 * D_INNER;
        const float* Wdt = W_dt   + (size_t)i * D_INNER * DT_RANK;
        const float* Al  = A_log  + (size_t)i * D_INNER * D_STATE;
        const float* Dpl = Dp     + (size_t)i * D_INNER;
        const float* Wo  = W_out  + (size_t)i * D_MODEL * D_INNER;
        const float* nw  = norm_w + (size_t)i * D_MODEL;

        rmsnorm_kernel<<<BL, blk, 0, stream>>>(x, nw, xn);
        // res = xn @ Wr^T ; xi = xn @ Wl^T   (2048 x 2048 x 1024)
        gemm_wmma_f32<0, 4, 4><<<grd44(BL, D_INNER), blk, 0, stream>>>(xn, Wr, nullptr, res,
                                                                       BL, D_INNER, D_MODEL, D_MODEL, D_INNER);
        gemm_wmma_f32<0, 4, 4><<<grd44(BL, D_INNER), blk, 0, stream>>>(xn, Wl, nullptr, xi,
                                                                       BL, D_INNER, D_MODEL, D_MODEL, D_INNER);
        conv_silu_kernel<<<(BL * D_INNER) / 256, blk, 0, stream>>>(xi, cw, cb, u);
        // x_dbl = u @ Wx^T  (N = 96 = DT_RANK + 2*D_STATE); 6 full 16-tiles
        gemm_wmma_f32<0, 4, 2><<<grd42(BL, 96), blk, 0, stream>>>(u, Wx, nullptr, xdbl,
                                                                  BL, 96, D_INNER, D_INNER, 96);
        // delta = softplus(dlt @ Wdt^T); dlt = x_dbl[:, :64] (lda = 96)
        gemm_wmma_f32<1, 4, 4><<<grd44(BL, D_INNER), blk, 0, stream>>>(xdbl, Wdt, nullptr, delta,
                                                                       BL, D_INNER, DT_RANK, 96, D_INNER);
        scan_kernel<<<dim3(D_INNER / 256, BB), blk, 0, stream>>>(u, delta, xdbl, Al, Dpl, res, yact);
        // x = yact @ Wo^T + x (residual epilogue; in-place safe: 1 lane/element)
        gemm_wmma_f32<2, 4, 4><<<grd44(BL, D_MODEL), blk, 0, stream>>>(yact, Wo, x, x,
                                                                       BL, D_MODEL, D_INNER, D_INNER, D_MODEL);
    }

    rmsnorm_kernel<<<BL, blk, 0, stream>>>(x, norm_f_w, xn);
    // logits = xn @ emb^T  (2048 x 32000 x 1024); last block's waves beyond N exit
    gemm_wmma_f32<0, 4, 4><<<grd44(BL, VOCAB), blk, 0, stream>>>(xn, emb, nullptr, out,
                                                                 BL, VOCAB, D_MODEL, D_MODEL, VOCAB);
}